// NewlinearRegression_60756607369622
// MI455X (gfx1250) — compile-verified
//
#include <hip/hip_runtime.h>
#include <stdint.h>

typedef __attribute__((ext_vector_type(2))) float v2f;
typedef __attribute__((ext_vector_type(8))) float v8f;

#define LDS_SLICE 1028  // 1024 + 4-float pad to de-conflict stride-SLICE accesses

// Low 32 bits of a flat pointer into __shared__ = LDS byte address
// (flat->LDS mapping truncates to addr[31:0], CDNA5 ISA).
__device__ __forceinline__ uint32_t lds_addr32(const void* p) {
  return (uint32_t)(uintptr_t)p;
}

// Async DMA global(16B/lane) -> LDS, tracked with ASYNCcnt (ISA §15.18.3 op 98).
__device__ __forceinline__ void async_load_b128(uint32_t lds_byte_addr,
                                                const void* gaddr) {
  asm volatile("global_load_async_to_lds_b128 %0, %1, off"
               :
               : "v"(lds_byte_addr), "v"((unsigned long long)(uintptr_t)gaddr)
               : "memory");
}

// Async DMA LDS -> global(16B/lane), tracked with ASYNCcnt (ISA op 102).
__device__ __forceinline__ void async_store_b128(void* gaddr,
                                                 uint32_t lds_byte_addr) {
  asm volatile("global_store_async_from_lds_b128 %0, %1, off"
               :
               : "v"((unsigned long long)(uintptr_t)gaddr), "v"(lds_byte_addr)
               : "memory");
}

__device__ __forceinline__ void wait_asynccnt0() {
  asm volatile("s_wait_asynccnt 0x0" ::: "memory");
}

// Preload this wave's A-fragments of W (32x32 row-major in LDS) for all
// 8 K-steps and both 16-row tiles: af[mi][kk]. ISA A-16x4-f32 layout:
// lane m=lane&15, half h=lane>>4 -> v0 = A[m][kk*4+2h], v1 = A[m][kk*4+2h+1].
// 32 VGPRs, loaded ONCE per step and reused by all 4 slice-GEMMs.
__device__ __forceinline__ void load_afrags(const float* __restrict__ Wl,
                                            int lane, v2f af[2][8]) {
  const int m = lane & 15;
  const int h = lane >> 4;
#pragma unroll
  for (int kk = 0; kk < 8; ++kk) {
    const int k0 = kk * 4 + 2 * h;
    af[0][kk].x = Wl[m * 32 + k0];
    af[0][kk].y = Wl[m * 32 + k0 + 1];
    af[1][kk].x = Wl[(m + 16) * 32 + k0];
    af[1][kk].y = Wl[(m + 16) * 32 + k0 + 1];
  }
}

// One-wave 32x32x32 fp32 GEMM, in place on an LDS operand:
//   S[0..31][0..31] (row stride rs)  <-  W * S, W given as preloaded A-frags.
// V_WMMA_F32_16X16X4_F32, 8 chained K=4 steps over a 2x2 grid of 16x16 tiles.
// ISA VGPR layouts (CDNA5 ISA):
//   B 4x16 : v0=B[2h][n], v1=B[2h+1][n], n=lane&15
//   C 16x16: vg = C[g+8h][n]
__device__ __forceinline__ void gemm32x32x32(const v2f af[2][8],
                                             float* __restrict__ S,
                                             int rs, int lane) {
  const int m = lane & 15;
  const int h = lane >> 4;
  v8f acc00 = {}, acc01 = {}, acc10 = {}, acc11 = {};
#pragma unroll
  for (int kk = 0; kk < 8; ++kk) {
    const int k0 = kk * 4 + 2 * h;
    v2f b0, b1;
    b0.x = S[k0 * rs + m];
    b0.y = S[(k0 + 1) * rs + m];
    b1.x = S[k0 * rs + m + 16];
    b1.y = S[(k0 + 1) * rs + m + 16];
    acc00 = __builtin_amdgcn_wmma_f32_16x16x4_f32(false, af[0][kk], false, b0,
                                                  (short)0, acc00, false, false);
    acc01 = __builtin_amdgcn_wmma_f32_16x16x4_f32(false, af[0][kk], false, b1,
                                                  (short)0, acc01, false, false);
    acc10 = __builtin_amdgcn_wmma_f32_16x16x4_f32(false, af[1][kk], false, b0,
                                                  (short)0, acc10, false, false);
    acc11 = __builtin_amdgcn_wmma_f32_16x16x4_f32(false, af[1][kk], false, b1,
                                                  (short)0, acc11, false, false);
  }
  // In-place writeback: all of this wave's DS loads precede these DS stores in
  // the in-order per-wave LDS stream, and each slice is owned by one wave.
#pragma unroll
  for (int g = 0; g < 8; ++g) {
    S[(8 * h + g) * rs + m]           = acc00[g];
    S[(8 * h + g) * rs + m + 16]      = acc01[g];
    S[(8 * h + g + 16) * rs + m]      = acc10[g];
    S[(8 * h + g + 16) * rs + m + 16] = acc11[g];
  }
}

// Fused two-mode Kronecker contraction over a 32x32x32-float tile.
// Tile element (a, r, c) lives at  src[base + a*strideA + r*strideR + c],
// with base = blockIdx.x * baseStride and c contiguous (32 floats).
// Step 1 contracts the r axis with Wr; step 2 contracts the a axis with Wa.
__global__ __launch_bounds__(256) void kron_pair_kernel(
    const float* __restrict__ src, float* __restrict__ dst,
    const float* __restrict__ Wr, const float* __restrict__ Wa,
    unsigned long long baseStride, unsigned long long strideA,
    unsigned long long strideR) {
  __shared__ float tile[32 * LDS_SLICE];
  __shared__ float wlds_r[32 * 32];
  __shared__ float wlds_a[32 * 32];

  const int t = threadIdx.x;     // 0..255 (8 wave32s)
  const int lane = t & 31;
  const int wave = t >> 5;
  const unsigned long long base = (unsigned long long)blockIdx.x * baseStride;

  // Kick off the async tile fill first: per 'a' slice, 1024 floats =
  // 256 lanes x 16B, DMA'd straight into LDS (no VGPR staging).
  const int lr = t >> 3;
  const int lc = (t & 7) * 4;
#pragma unroll 4
  for (int a = 0; a < 32; ++a) {
    async_load_b128(lds_addr32(&tile[a * LDS_SLICE + lr * 32 + lc]),
                    src + base + (unsigned long long)a * strideA +
                        (unsigned long long)lr * strideR + lc);
  }

  // Factor matrices into LDS (1024 floats each) while the DMA runs.
  ((float4*)wlds_r)[t] = ((const float4*)Wr)[t];
  ((float4*)wlds_a)[t] = ((const float4*)Wa)[t];

  wait_asynccnt0();   // this wave's async LDS writes are done
  __syncthreads();    // everyone's are

  // Step 1: for each slice a, tile[a][r'][c] = sum_r Wr[r'][r] * tile[a][r][c]
  {
    v2f af[2][8];
    load_afrags(wlds_r, lane, af);
#pragma unroll
    for (int q = 0; q < 4; ++q) {
      const int a = wave + 8 * q;
      gemm32x32x32(af, &tile[a * LDS_SLICE], 32, lane);
    }
  }
  __syncthreads();

  // Step 2: for each row r, tile[a'][r][c] = sum_a Wa[a'][a] * tile[a][r][c]
  {
    v2f af[2][8];
    load_afrags(wlds_a, lane, af);
#pragma unroll
    for (int q = 0; q < 4; ++q) {
      const int r = wave + 8 * q;
      gemm32x32x32(af, &tile[r * 32], LDS_SLICE, lane);
    }
  }
  __syncthreads();

  // Async DMA the finished tile LDS -> global, same addressing as the load.
#pragma unroll 4
  for (int a = 0; a < 32; ++a) {
    async_store_b128(dst + base + (unsigned long long)a * strideA +
                         (unsigned long long)lr * strideR + lc,
                     lds_addr32(&tile[a * LDS_SLICE + lr * 32 + lc]));
  }
  wait_asynccnt0();   // drain before S_ENDPGM (which also waits, belt+braces)
}

extern "C" void kernel_launch(void* const* d_in, const int* in_sizes, int n_in,
                              void* d_out, int out_size, void* d_ws,
                              size_t ws_size, hipStream_t stream) {
  (void)in_sizes; (void)n_in; (void)out_size; (void)ws_size;
  const float* x  = (const float*)d_in[0];
  const float* W0 = (const float*)d_in[1];
  const float* W1 = (const float*)d_in[2];
  const float* W2 = (const float*)d_in[3];
  const float* W3 = (const float*)d_in[4];
  float* out = (float*)d_out;
  float* tmp = (float*)d_ws;  // 32^5 floats = 128 MiB intermediate (fits in L2)

  // t[i0,i1,i2,i3,b]: strides (elements) 1048576 / 32768 / 1024 / 32 / 1.
  // Pass 1: per (i0,i1) block (base = blk*32768, contiguous 128 KiB tile):
  //   contract i3 with W3 (a=i2 slices), then i2 with W2.
  kron_pair_kernel<<<1024, 256, 0, stream>>>(x, tmp, W3, W2,
                                             32768ULL, 1024ULL, 32ULL);
  // Pass 2: per (i2, chunk) block (base = blk*32, rows of 32 contiguous):
  //   contract i1 with W1 (a=i0 slices), then i0 with W0.
  kron_pair_kernel<<<1024, 256, 0, stream>>>(tmp, out, W1, W0,
                                             32ULL, 1048576ULL, 32768ULL);
}